// ModelNew_10522669875254
// MI455X (gfx1250) — compile-verified
//
#include <hip/hip_runtime.h>
#include <hip/hip_bf16.h>
#include <math.h>

#ifndef __has_builtin
#define __has_builtin(x) 0
#endif

#if __has_builtin(__builtin_amdgcn_global_load_async_to_lds_b32)
#define HAVE_ASYNC_LDS 1
#else
#define HAVE_ASYNC_LDS 0
#endif

#define BLOCK_THREADS 256
#define WAVES_PER_BLOCK (BLOCK_THREADS / 32)
#define ROW_CAP 512  // floats of LDS staging per wave (covers max expected degree ~230)

typedef __attribute__((address_space(1))) int as1_int;  // "__device__ int"
typedef __attribute__((address_space(3))) int as3_int;  // "__shared__ int"

__device__ __forceinline__ void wait_async_zero() {
#if __has_builtin(__builtin_amdgcn_s_wait_asynccnt)
    __builtin_amdgcn_s_wait_asynccnt(0);
    __asm__ __volatile__("" ::: "memory");
#else
    __asm__ __volatile__("s_wait_asynccnt 0" ::: "memory");
#endif
}

__device__ __forceinline__ float wave_max_f32(float v) {
    #pragma unroll
    for (int off = 16; off > 0; off >>= 1)
        v = fmaxf(v, __shfl_xor(v, off, 32));
    return v;
}

__device__ __forceinline__ float wave_sum_f32(float v) {
    #pragma unroll
    for (int off = 16; off > 0; off >>= 1)
        v += __shfl_xor(v, off, 32);
    return v;
}

__global__ __launch_bounds__(BLOCK_THREADS) void seg_softmax_csr_kernel(
    const int* __restrict__ row_ptr,
    const float* __restrict__ scores,
    float* __restrict__ out,
    int num_nodes)
{
    __shared__ float sbuf[WAVES_PER_BLOCK][ROW_CAP];

    const int lane  = threadIdx.x & 31;
    const int wslot = threadIdx.x >> 5;
    // wave-uniform row index so row_ptr loads scalarize (s_load)
    const int wave0  = __builtin_amdgcn_readfirstlane(
                           (int)(blockIdx.x * WAVES_PER_BLOCK + wslot));
    const int nwaves = (int)(gridDim.x * WAVES_PER_BLOCK);
    float* buf = sbuf[wslot];

    for (int row = wave0; row < num_nodes; row += nwaves) {
        const int s   = row_ptr[row];
        const int e   = row_ptr[row + 1];
        const int len = e - s;
        if (len <= 0) continue;

        // Prefetch next row's score data (gfx1250 global_prefetch_b8).
        const int next_row = row + nwaves;
        if (next_row < num_nodes) {
            const int ns = row_ptr[next_row];
            const int ne = row_ptr[next_row + 1];
            const int noff = lane * 16;  // 64B cacheline stride per lane
            if (ns + noff < ne)
                __builtin_prefetch(scores + ns + noff, 0, 3);
        }

        if (len <= 32) {
            // ---- Fast path A (~86% of rows): one element per lane, registers only ----
            const float x = (lane < len) ? scores[s + lane] : -INFINITY;
            const float m = wave_max_f32(x);
            const float v = (lane < len) ? __expf(x - m) : 0.0f;
            const float denom = wave_sum_f32(v);
            if (lane < len) out[s + lane] = v / denom;
        } else if (len <= 64) {
            // ---- Fast path B (cum. ~98% of rows): two elements per lane ----
            const int  i0 = s + lane;
            const int  i1 = s + 32 + lane;
            const bool p1 = (32 + lane) < len;
            const float x0 = scores[i0];  // always valid: len > 32
            const float x1 = p1 ? scores[i1] : -INFINITY;
            const float m  = wave_max_f32(fmaxf(x0, x1));
            const float v0 = __expf(x0 - m);
            const float v1 = p1 ? __expf(x1 - m) : 0.0f;
            const float denom = wave_sum_f32(v0 + v1);
            out[i0] = v0 / denom;
            if (p1) out[i1] = v1 / denom;
        } else if (len <= ROW_CAP) {
            // ---- LDS path: async global -> LDS staging (ASYNCcnt), 3 passes ----
            for (int i = lane; i < len; i += 32) {
#if HAVE_ASYNC_LDS
                __builtin_amdgcn_global_load_async_to_lds_b32(
                    (as1_int*)(scores + s + i),
                    (as3_int*)(buf + i),
                    0 /*imm offset*/, 0 /*cpol*/);
#else
                buf[i] = scores[s + i];
#endif
            }
            wait_async_zero();

            float m = -INFINITY;
            for (int i = lane; i < len; i += 32)
                m = fmaxf(m, buf[i]);
            m = wave_max_f32(m);

            float partial = 0.0f;
            for (int i = lane; i < len; i += 32) {
                const float v = __expf(buf[i] - m);
                buf[i] = v;
                partial += v;
            }
            const float denom = wave_sum_f32(partial);

            for (int i = lane; i < len; i += 32)
                out[s + i] = buf[i] / denom;
        } else {
            // ---- Rare long-row fallback: direct global (re-reads hit 192MB L2) ----
            float m = -INFINITY;
            for (int i = lane; i < len; i += 32)
                m = fmaxf(m, scores[s + i]);
            m = wave_max_f32(m);

            float partial = 0.0f;
            for (int i = lane; i < len; i += 32)
                partial += __expf(scores[s + i] - m);
            const float denom = wave_sum_f32(partial);

            for (int i = lane; i < len; i += 32)
                out[s + i] = __expf(scores[s + i] - m) / denom;
        }
    }
}

extern "C" void kernel_launch(void* const* d_in, const int* in_sizes, int n_in,
                              void* d_out, int out_size, void* d_ws, size_t ws_size,
                              hipStream_t stream) {
    const int*   row_ptr = (const int*)d_in[0];     // [N_NODES + 1], int32
    const float* scores  = (const float*)d_in[1];   // [N_EDGES], fp32
    float*       out     = (float*)d_out;           // [N_EDGES], fp32

    const int num_nodes = in_sizes[0] - 1;

    // ~32K waves: enough parallelism to cover both HBM latency and issue time.
    const int blocks = 4096;
    seg_softmax_csr_kernel<<<blocks, BLOCK_THREADS, 0, stream>>>(
        row_ptr, scores, out, num_nodes);
}